// myPCANorm_38560216383955
// MI455X (gfx1250) — compile-verified
//
#include <hip/hip_runtime.h>

// ---------------------------------------------------------------------------
// PCANorm on gfx1250 (MI455X). f32 path using V_WMMA_F32_16X16X4_F32.
//   K1: per-channel mean / rstd (ddof=1)
//   K2: xxt = xf xf^T / m + eps I   (LDS-staged SYRK, 8-wave K-split, WMMA f32)
//   K3: deflated power iteration scan (basis init; lambda = ||A v||; trace mask)
//   K4: P = (keep*V)^T V            (WMMA f32)
//   K5: out = (P @ xf) * w + b      (WMMA f32, xf normalized on the fly)
// Workspace: 148032 floats (~592 KB).
// ---------------------------------------------------------------------------

#define C_DIM 256
#define N_IMG 32
#define HW    4096
#define CHW   (C_DIM * HW)          // 1048576
#define M_TOT (N_IMG * HW)          // 131072
#define EPSV  1e-5f
#define N_EIG 64
#define N_PIT 19
#define ETHRESH 0.95f

typedef __attribute__((ext_vector_type(2))) float v2f;
typedef __attribute__((ext_vector_type(8))) float v8f;

// ws layout (float offsets)
#define WS_MU    0
#define WS_RSTD  256
#define WS_XXT   512
#define WS_V     (WS_XXT + C_DIM * C_DIM)          // 66048
#define WS_KEEP  (WS_V + N_EIG * C_DIM)            // 82432
#define WS_P     (WS_KEEP + 64)                    // 82496 (+65536 -> 148032 floats)

// ---------------------------------------------------------------- K1: stats
__global__ void k_stats(const float* __restrict__ x, float* __restrict__ ws) {
  const int c = blockIdx.x, tid = threadIdx.x;
  float s = 0.f, ss = 0.f;
  for (int n = 0; n < N_IMG; ++n) {
    const float* p = x + (size_t)n * CHW + (size_t)c * HW;
    for (int i = tid; i < HW; i += 256) { float v = p[i]; s += v; ss += v * v; }
  }
  __shared__ float r1[256], r2[256];
  r1[tid] = s; r2[tid] = ss; __syncthreads();
  for (int st = 128; st > 0; st >>= 1) {
    if (tid < st) { r1[tid] += r1[tid + st]; r2[tid] += r2[tid + st]; }
    __syncthreads();
  }
  if (tid == 0) {
    float mu  = r1[0] / (float)M_TOT;
    float var = (r2[0] - r1[0] * mu) / (float)(M_TOT - 1);   // ddof = 1
    ws[WS_MU + c]   = mu;
    ws[WS_RSTD + c] = rsqrtf(var + EPSV);
  }
}

// ------------------------------------------------------------- K2: xxt SYRK
// One block per 16x16 tile of xxt. 8 waves split K; LDS staging + reduce.
#define KC  256                 // K chunk staged in LDS
#define KP  260                 // padded row (floats) to avoid bank conflicts
__global__ void k_xxt(const float* __restrict__ x, float* __restrict__ ws) {
  __shared__ float sA[16 * KP];
  __shared__ float sB[16 * KP];
  __shared__ float sRed[8][32][8];
  __shared__ float muA[16], rsA[16], muB[16], rsB[16];

  const int tid = threadIdx.x;
  const int bi  = blockIdx.x;
  const int cA  = (bi >> 4) * 16, cB = (bi & 15) * 16;
  if (tid < 16)      { muA[tid] = ws[WS_MU + cA + tid];      rsA[tid] = ws[WS_RSTD + cA + tid]; }
  else if (tid < 32) { int t = tid - 16;
                       muB[t]   = ws[WS_MU + cB + t];        rsB[t]   = ws[WS_RSTD + cB + t]; }
  __syncthreads();

  const int w = tid >> 5, lane = tid & 31, l = lane & 15, h = lane >> 4;
  v8f acc = {0.f, 0.f, 0.f, 0.f, 0.f, 0.f, 0.f, 0.f};

  const int NCHUNK = M_TOT / KC;                   // 512
  for (int ch = 0; ch < NCHUNK; ++ch) {
    const int n  = ch >> 4;                        // 16 chunks per image
    const int i0 = (ch & 15) * KC;
    const float* base = x + (size_t)n * CHW + i0;

    // Speculative prefetch of next chunk (global_prefetch_b8). 128B lines.
    if (ch + 1 < NCHUNK) {
      const int nn = (ch + 1) >> 4, ii = ((ch + 1) & 15) * KC;
      const float* nb = x + (size_t)nn * CHW + ii;
      const int pr = tid & 127, prow = pr >> 3, poff = (pr & 7) * 32;
      const float* pb = (tid < 128) ? (nb + (size_t)(cA + prow) * HW + poff)
                                    : (nb + (size_t)(cB + prow) * HW + poff);
      __builtin_prefetch(pb, 0, 0);
    }

    // Cooperative coalesced stage: 16 rows x 256 floats, normalized.
    for (int it = 0; it < 4; ++it) {
      const int lin = it * 256 + tid;              // 0..1023
      const int row = lin >> 6, c4 = lin & 63;
      float4 va = *(const float4*)(base + (size_t)(cA + row) * HW + c4 * 4);
      float m_ = muA[row], r_ = rsA[row];
      va.x = (va.x - m_) * r_; va.y = (va.y - m_) * r_;
      va.z = (va.z - m_) * r_; va.w = (va.w - m_) * r_;
      *(float4*)(&sA[row * KP + c4 * 4]) = va;
      float4 vb = *(const float4*)(base + (size_t)(cB + row) * HW + c4 * 4);
      m_ = muB[row]; r_ = rsB[row];
      vb.x = (vb.x - m_) * r_; vb.y = (vb.y - m_) * r_;
      vb.z = (vb.z - m_) * r_; vb.w = (vb.w - m_) * r_;
      *(float4*)(&sB[row * KP + c4 * 4]) = vb;
    }
    __syncthreads();

    // Wave w handles K-slice [w*32, w*32+32).
    const int kb = w * 32;
#pragma unroll
    for (int kk = 0; kk < 32; kk += 4) {
      const int ko = kb + kk + 2 * h;              // VGPR0: K=2h, VGPR1: K=2h+1
      v2f a = *(const v2f*)(&sA[l * KP + ko]);
      v2f b = *(const v2f*)(&sB[l * KP + ko]);
      acc = __builtin_amdgcn_wmma_f32_16x16x4_f32(false, a, false, b,
                                                  (short)0, acc, false, false);
    }
    __syncthreads();
  }

  // Cross-wave K reduce, then scale + eps*I, write tile.
  for (int r = 0; r < 8; ++r) sRed[w][lane][r] = acc[r];
  __syncthreads();
  if (tid < 32) {
    const int ll = tid & 15, hh = tid >> 4;
    for (int r = 0; r < 8; ++r) {
      float s = 0.f;
      for (int ww = 0; ww < 8; ++ww) s += sRed[ww][tid][r];
      const int row = cA + r + 8 * hh, col = cB + ll;
      ws[WS_XXT + row * C_DIM + col] =
          s / (float)M_TOT + ((row == col) ? EPSV : 0.f);
    }
  }
}

// ----------------------------------------------- K3: deflated power-iter scan
__global__ void k_scan(float* __restrict__ ws) {
  __shared__ float v[256], red[256];
  __shared__ float sCum, sTrace, sLam;
  const int tid = threadIdx.x;
  float* A = ws + WS_XXT;

  // trace(xxt) == sum of all eigenvalues (exact substitute for sum(e))
  red[tid] = A[tid * C_DIM + tid]; __syncthreads();
  for (int s = 128; s > 0; s >>= 1) { if (tid < s) red[tid] += red[tid + s]; __syncthreads(); }
  if (tid == 0) { sTrace = red[0]; sCum = 0.f; }
  __syncthreads();

  for (int k = 0; k < N_EIG; ++k) {
    v[tid] = (tid == k) ? 1.f : 0.f;               // basis init (SVD surrogate)
    __syncthreads();

    for (int it = 0; it < N_PIT; ++it) {
      float accv = 0.f;
      const float* Ar = A + tid * C_DIM;
      for (int j = 0; j < C_DIM; j += 4) {
        float4 a4 = *(const float4*)(Ar + j);
        accv += a4.x * v[j] + a4.y * v[j + 1] + a4.z * v[j + 2] + a4.w * v[j + 3];
      }
      red[tid] = accv * accv; __syncthreads();
      for (int s = 128; s > 0; s >>= 1) { if (tid < s) red[tid] += red[tid + s]; __syncthreads(); }
      const float rn = rsqrtf(red[0] + 1e-38f);
      __syncthreads();
      v[tid] = accv * rn;
      __syncthreads();
    }

    // w = A v (deflation vector); lambda ~= ||A v||
    float accv = 0.f;
    {
      const float* Ar = A + tid * C_DIM;
      for (int j = 0; j < C_DIM; j += 4) {
        float4 a4 = *(const float4*)(Ar + j);
        accv += a4.x * v[j] + a4.y * v[j + 1] + a4.z * v[j + 2] + a4.w * v[j + 3];
      }
    }
    red[tid] = accv * accv; __syncthreads();
    for (int s = 128; s > 0; s >>= 1) { if (tid < s) red[tid] += red[tid + s]; __syncthreads(); }
    if (tid == 0) sLam = sqrtf(red[0]);
    __syncthreads();

    ws[WS_V + k * C_DIM + tid] = v[tid];
    if (tid == 0) {
      ws[WS_KEEP + k] = (k == 0 || (sCum / sTrace) < ETHRESH) ? 1.f : 0.f;
      sCum += sLam;
    }
    // A -= w v^T  (each thread owns one row)
    float* Aw = A + tid * C_DIM;
    for (int j = 0; j < C_DIM; ++j) Aw[j] -= accv * v[j];
    __syncthreads();
  }
}

// ----------------------------------------------------- K4: P = (keep*V)^T V
__global__ void k_buildP(float* __restrict__ ws) {
  const int tid = threadIdx.x;
  const int w = tid >> 5, lane = tid & 31, l = lane & 15, h = lane >> 4;
  const int t = blockIdx.x * 8 + w;                // 256 tiles total
  const int cI = (t >> 4) * 16, cJ = (t & 15) * 16;
  const float* V  = ws + WS_V;
  const float* kp = ws + WS_KEEP;

  v8f acc = {0.f, 0.f, 0.f, 0.f, 0.f, 0.f, 0.f, 0.f};
  for (int k0 = 0; k0 < N_EIG; k0 += 4) {
    const int kr = k0 + 2 * h;
    v2f a, b;
    a.x = V[kr * C_DIM + cI + l] * kp[kr];         // A = Vm^T : A[i,k]
    a.y = V[(kr + 1) * C_DIM + cI + l] * kp[kr + 1];
    b.x = V[kr * C_DIM + cJ + l];                  // B = V    : B[k,j]
    b.y = V[(kr + 1) * C_DIM + cJ + l];
    acc = __builtin_amdgcn_wmma_f32_16x16x4_f32(false, a, false, b,
                                                (short)0, acc, false, false);
  }
  for (int r = 0; r < 8; ++r) {
    const int row = cI + r + 8 * h, col = cJ + l;
    ws[WS_P + row * C_DIM + col] = acc[r];
  }
}

// ------------------------------------------- K5: out = (P @ xf) * w + b
// grid (1024, 16): y-dim picks 16 rows of P (staged in LDS), 8 waves pick j-tiles.
__global__ void k_out(const float* __restrict__ x, const float* __restrict__ weight,
                      const float* __restrict__ bias, const float* __restrict__ ws,
                      float* __restrict__ out) {
  __shared__ float sP[16 * KP];
  __shared__ float sMu[256], sRs[256], sW[16], sBi[16];
  const int tid = threadIdx.x;
  const int cI  = blockIdx.y * 16;

  for (int it = 0; it < 16; ++it) {
    const int lin = it * 256 + tid, row = lin >> 8, col = lin & 255;
    sP[row * KP + col] = ws[WS_P + (cI + row) * C_DIM + col];
  }
  sMu[tid] = ws[WS_MU + tid];
  sRs[tid] = ws[WS_RSTD + tid];
  if (tid < 16) { sW[tid] = weight[cI + tid]; sBi[tid] = bias[cI + tid]; }
  __syncthreads();

  const int w = tid >> 5, lane = tid & 31, l = lane & 15, h = lane >> 4;
  const int j0 = (blockIdx.x * 8 + w) * 16;        // never crosses an image (4096%16==0)
  const int n = j0 >> 12, i0 = j0 & 4095;
  const float* xb = x + (size_t)n * CHW + i0;

  v8f acc = {0.f, 0.f, 0.f, 0.f, 0.f, 0.f, 0.f, 0.f};
  for (int k0 = 0; k0 < C_DIM; k0 += 4) {
    const int kr = k0 + 2 * h;
    v2f a = *(const v2f*)(&sP[l * KP + kr]);       // A = P tile rows
    v2f b;                                         // B = xf[k, j0+l], coalesced in l
    b.x = (xb[(size_t)kr * HW + l]       - sMu[kr])     * sRs[kr];
    b.y = (xb[(size_t)(kr + 1) * HW + l] - sMu[kr + 1]) * sRs[kr + 1];
    acc = __builtin_amdgcn_wmma_f32_16x16x4_f32(false, a, false, b,
                                                (short)0, acc, false, false);
  }
  for (int r = 0; r < 8; ++r) {
    const int c = cI + r + 8 * h;
    const float y = acc[r] * sW[r + 8 * h] + sBi[r + 8 * h];
    out[(size_t)n * CHW + (size_t)c * HW + i0 + l] = y;
  }
}

// ---------------------------------------------------------------------------
extern "C" void kernel_launch(void* const* d_in, const int* in_sizes, int n_in,
                              void* d_out, int out_size, void* d_ws, size_t ws_size,
                              hipStream_t stream) {
  (void)in_sizes; (void)n_in; (void)out_size; (void)ws_size;
  const float* x      = (const float*)d_in[0];
  const float* weight = (const float*)d_in[1];
  const float* bias   = (const float*)d_in[2];
  float* out = (float*)d_out;
  float* ws  = (float*)d_ws;   // needs 148032 floats (~592 KB)

  k_stats <<<256, 256, 0, stream>>>(x, ws);
  k_xxt   <<<256, 256, 0, stream>>>(x, ws);
  k_scan  <<<1,   256, 0, stream>>>(ws);
  k_buildP<<<32,  256, 0, stream>>>(ws);
  dim3 g5(1024, 16);
  k_out   <<<g5,  256, 0, stream>>>(x, weight, bias, ws, out);
}